// DirectVoxGO_33938831573255
// MI455X (gfx1250) — compile-verified
//
#include <hip/hip_runtime.h>

#define N_LANES 32
#define WAVES_PER_BLOCK 8
#define BLOCK_THREADS (N_LANES * WAVES_PER_BLOCK)

// ---- CDNA5 async global->LDS copy (ASYNCcnt-tracked), inline asm form ------
// Per-lane: copies 4 bytes from the global address in ga to the LDS address in lds_addr.
// lds_escape makes the shared buffer visibly written-by-asm so its loads
// cannot be optimized away.
__device__ __forceinline__ void async_load_f32_to_lds(unsigned lds_addr,
                                                      const float* gptr,
                                                      void* lds_escape) {
    unsigned long long ga = (unsigned long long)(size_t)gptr;
    asm volatile("global_load_async_to_lds_b32 %0, %1, off"
                 :
                 : "v"(lds_addr), "v"(ga), "r"(lds_escape)
                 : "memory");
}

#define ASYNC_WAIT_1() asm volatile("s_wait_asynccnt 0x1" ::: "memory")
#define ASYNC_WAIT_0() asm volatile("s_wait_asynccnt 0x0" ::: "memory")

// numerically stable softplus matching jax.nn.softplus = max(x,0)+log1p(exp(-|x|))
__device__ __forceinline__ float softplus_f32(float x) {
    return fmaxf(x, 0.0f) + log1pf(expf(-fabsf(x)));
}

__global__ __launch_bounds__(BLOCK_THREADS)
void dvgo_alpha2weights_kernel(const float* __restrict__ density,
                               const float* __restrict__ p_shift,
                               const float* __restrict__ p_interval,
                               const int*   __restrict__ ray_id,
                               int n_pts, int n_rays,
                               float* __restrict__ weights,
                               float* __restrict__ alphainv_last) {
    __shared__ float stage[WAVES_PER_BLOCK][2][N_LANES];

    const int lane = (int)(threadIdx.x & (N_LANES - 1));
    const int wv   = (int)(threadIdx.x >> 5);

    // Ray index is wave-uniform by construction (tid>>5); pin it with
    // readfirstlane so the compiler's uniformity analysis scalarizes the
    // binary search (SALU + s_load probes) and the chunk-loop control flow.
    const int r = __builtin_amdgcn_readfirstlane(
        (int)((blockIdx.x * (unsigned)BLOCK_THREADS + threadIdx.x) >> 5));
    if (r >= n_rays) return;   // wave-uniform exit

    const float shift    = *p_shift;
    const float interval = *p_interval;

    // Fused dual lower_bound over sorted ray_id: start = first idx with id>=r,
    // end = first idx with id>=r+1. Scalar addresses -> SMEM probes, all
    // L2-resident; the two independent chains interleave for latency overlap.
    int s_lo = 0, s_hi = n_pts;
    int e_lo = 0, e_hi = n_pts;
    while ((s_lo < s_hi) || (e_lo < e_hi)) {
        if (s_lo < s_hi) {
            int m = (s_lo + s_hi) >> 1;
            if (ray_id[m] < r) s_lo = m + 1; else s_hi = m;
        }
        if (e_lo < e_hi) {
            int m = (e_lo + e_hi) >> 1;
            if (ray_id[m] < r + 1) e_lo = m + 1; else e_hi = m;
        }
    }
    const int start   = __builtin_amdgcn_readfirstlane(s_lo);
    const int end     = __builtin_amdgcn_readfirstlane(e_lo);
    const int nchunks = (end - start + N_LANES - 1) >> 5;

    // Per-lane LDS staging addresses (double buffer, 128B apart).
    void*    lds_ptr  = (void*)&stage[wv][0][0];
    unsigned lds_base = (unsigned)(size_t)&stage[wv][0][lane];

    // Prologue: issue chunk 0 (addresses clamped into the array; tail lanes
    // load harmless in-bounds data and are masked at compute time).
    if (nchunks > 0) {
        int i0 = start + lane; if (i0 > n_pts - 1) i0 = n_pts - 1;
        async_load_f32_to_lds(lds_base, density + i0, lds_ptr);
    }

    float carry = 0.0f;  // running sum of log(1-alpha) within this ray
    for (int c = 0; c < nchunks; ++c) {
        if (c + 1 < nchunks) {
            int i1 = start + (c + 1) * N_LANES + lane;
            if (i1 > n_pts - 1) i1 = n_pts - 1;
            async_load_f32_to_lds(lds_base + (unsigned)(((c + 1) & 1) << 7),
                                  density + i1, lds_ptr);
            ASYNC_WAIT_1();   // in-order completion => chunk c is in LDS
        } else {
            ASYNC_WAIT_0();
        }

        const int i = start + c * N_LANES + lane;
        float l = 0.0f;      // log(1 - alpha), 0 for masked lanes
        if (i < end) {
            float x = stage[wv][c & 1][lane] + shift;
            l = -interval * softplus_f32(x);
        }

        // Hillis-Steele inclusive scan across the wave (wave32).
        float incl = l;
        #pragma unroll
        for (int off = 1; off < N_LANES; off <<= 1) {
            float t = __shfl_up(incl, off, N_LANES);
            if (lane >= off) incl += t;
        }

        if (i < end) {
            float T     = expf(carry + (incl - l));  // transmittance before sample
            float alpha = 1.0f - expf(l);
            weights[i]  = alpha * T;
        }
        carry += __shfl(incl, N_LANES - 1, N_LANES); // chunk total -> all lanes
    }

    if (lane == 0) alphainv_last[r] = expf(carry);   // empty ray -> exp(0)=1
}

extern "C" void kernel_launch(void* const* d_in, const int* in_sizes, int n_in,
                              void* d_out, int out_size, void* d_ws, size_t ws_size,
                              hipStream_t stream) {
    (void)n_in; (void)d_ws; (void)ws_size;
    const float* density    = (const float*)d_in[0];
    const float* p_shift    = (const float*)d_in[1];
    const float* p_interval = (const float*)d_in[2];
    const int*   ray_id     = (const int*)d_in[3];

    const int n_pts  = in_sizes[0];
    const int n_rays = out_size - n_pts;   // weights || alphainv_last

    float* weights  = (float*)d_out;
    float* alphainv = weights + n_pts;

    dim3 block(BLOCK_THREADS);
    dim3 grid((unsigned)((n_rays + WAVES_PER_BLOCK - 1) / WAVES_PER_BLOCK));
    dvgo_alpha2weights_kernel<<<grid, block, 0, stream>>>(
        density, p_shift, p_interval, ray_id, n_pts, n_rays, weights, alphainv);
}